// ATrousWaveletDecomposition_73383811219860
// MI455X (gfx1250) — compile-verified
//
#include <hip/hip_runtime.h>
#include <hip/hip_bf16.h>

// A-trous wavelet bank, MI455X (gfx1250, wave32).
//
// Strategy: each dilated cumsum kernel is piecewise-constant, so the conv
// output is an affine combo of 5 clamped prefix-sum lookups. This turns a
// 12.9 GFLOP direct conv into a pure-bandwidth problem (151 MB of stores,
// ~6.5us at 23.3 TB/s HBM). Per block: TDM the row into LDS, WMMA-based
// block scan to build the exclusive prefix P[0..16384], then stream outputs
// with non-temporal stores.

#define SLEN    16384
#define NSCALES 9
#define OUTLEN  16383          // min_len across scales (scale 0 has even tap count)
#define NTILE   64             // 64 tiles of 256 elements per row
#define BLOCK   1024           // 32 waves

#ifndef __has_builtin
#define __has_builtin(x) 0
#endif

#if defined(__HIP_DEVICE_COMPILE__) && __has_builtin(__builtin_amdgcn_tensor_load_to_lds)
#define USE_TDM 1
#else
#define USE_TDM 0
#endif

#if defined(__HIP_DEVICE_COMPILE__) && __has_builtin(__builtin_amdgcn_wmma_f32_16x16x4_f32)
#define USE_WMMA 1
#else
#define USE_WMMA 0
#endif

typedef float        v2f __attribute__((ext_vector_type(2)));
typedef float        v8f __attribute__((ext_vector_type(8)));
typedef unsigned int v4u __attribute__((ext_vector_type(4)));
typedef int          v4i __attribute__((ext_vector_type(4)));
typedef int          v8i __attribute__((ext_vector_type(8)));

__device__ __forceinline__ int clampP(int v) {
    v = v < 0 ? 0 : v;
    return v > SLEN ? SLEN : v;
}

__global__ __launch_bounds__(BLOCK)
void atrous_prefix_kernel(const float* __restrict__ x, float* __restrict__ out) {
    // P[0..16384]: exclusive prefix sums of the row (P[0]=0).  TDM deposits the
    // raw row at P[1..16384]; the scan then runs in place (each tile is private
    // to one wave, and a wave's LDS loads are issued before its stores).
    __shared__ float sP[SLEN + 1 + 14];   // +14 pad floats (bank stagger / align)
    __shared__ float sTot[NTILE];

    const int row  = blockIdx.x;
    const int tid  = threadIdx.x;
    const int lane = tid & 31;
    const int wave = tid >> 5;
    const int h    = lane >> 4;   // lane half (K split for A/B operands)
    const int m    = lane & 15;   // M for A, N for B and C/D

    // ---- Stage 1: row -> LDS -----------------------------------------------
#if USE_TDM
    if (tid == 0) {
        unsigned long long ga = (unsigned long long)(const void*)(x + (size_t)row * SLEN);
        unsigned lds_off = (unsigned)(unsigned long long)(void*)&sP[1]; // addr[31:0] = LDS byte offset
        // D# group0: count=1(valid), lds_addr, 57-bit global_addr, type=2
        v4u g0;
        g0.x = 1u;
        g0.y = lds_off;
        g0.z = (unsigned)ga;
        g0.w = (unsigned)((ga >> 32) & 0x01FFFFFFull) | (2u << 30);
        // D# group1: data_size=4B; tensor_dim0=16384; tensor_dim1=1;
        // tile_dim0=16384; tile_dim1=1; tensor_dim0_stride=16384
        v8i g1;
        g1[0] = 2 << 16;                 // [17:16] data_size = 2 (4 bytes)
        g1[1] = (SLEN & 0xFFFF) << 16;   // tensor_dim0[15:0]  in bits[63:48]
        g1[2] = ((SLEN >> 16) & 0xFFFF) | (1 << 16); // dim0 hi | tensor_dim1=1 lo
        g1[3] = (SLEN & 0xFFFF) << 16;   // tensor_dim1 hi=0 | tile_dim0 = 16384
        g1[4] = 1;                       // tile_dim1 = 1, tile_dim2 = 0
        g1[5] = SLEN;                    // tensor_dim0_stride[31:0]
        g1[6] = 0;
        g1[7] = 0;
        v4i gz  = {0, 0, 0, 0};          // groups 2/3 unused (<=2D tensor)
        v8i gz8 = {0, 0, 0, 0, 0, 0, 0, 0};
        // clang-23 / therock-10.0 toolchain: 6-arg form
        // (g0, g1, g2, g3, <extra v8i, zero-filled per probe>, cpol)
        __builtin_amdgcn_tensor_load_to_lds(g0, g1, gz, gz, gz8, 0);
        __builtin_amdgcn_s_wait_tensorcnt(0);
        sP[0] = 0.0f;
    }
#else
    {
        const float* xr = x + (size_t)row * SLEN;
        for (int e = tid; e < SLEN; e += BLOCK) sP[e + 1] = xr[e];
        if (tid == 0) sP[0] = 0.0f;
    }
#endif
    __syncthreads();

    // ---- Stage 2: per-wave tile scans (256 elements = 16 segs x 16) --------
    // Scan-by-matmul: D = X * U with U upper-triangular ones, K-split into
    // four 16x16x4 f32 WMMAs.  Row r of D = inclusive scan of segment r.
    for (int tt = 0; tt < 2; ++tt) {
        const int t    = wave * 2 + tt;
        const int base = t * 256 + 1;
#if USE_WMMA
        v2f a[4], b[4];
#pragma unroll
        for (int i = 0; i < 4; ++i) {
            // A (16x4 f32): lanes 0-15 K={0,1}, lanes 16-31 K={2,3}; M=lane%16
            const int k0 = 4 * i + 2 * h;
            a[i].x = sP[base + m * 16 + k0];
            a[i].y = sP[base + m * 16 + k0 + 1];
            // B (4x16 f32): K = vgpr + 2*laneHalf, N = lane%16 (inferred from
            // the documented 8/16-bit B layouts). U[k][n] = (k <= n).
            b[i].x = (k0     <= m) ? 1.0f : 0.0f;
            b[i].y = (k0 + 1 <= m) ? 1.0f : 0.0f;
        }
        v8f d = {};
#pragma unroll
        for (int i = 0; i < 4; ++i)
            d = __builtin_amdgcn_wmma_f32_16x16x4_f32(false, a[i], false, b[i],
                                                      (short)0, d, false, false);
        // Segment totals D[q][15] -> exclusive offsets (uniform per wave).
        float seg[16];
#pragma unroll
        for (int r = 0; r < 8; ++r) {
            seg[r]     = __shfl(d[r], 15);
            seg[r + 8] = __shfl(d[r], 31);
        }
        float off[16];
        off[0] = 0.0f;
#pragma unroll
        for (int q = 1; q < 16; ++q) off[q] = off[q - 1] + seg[q - 1];
        // D layout: vgpr r, lanes 0-15 -> row r, lanes 16-31 -> row r+8; col = lane%16
#pragma unroll
        for (int r = 0; r < 8; ++r)
            sP[base + (r + 8 * h) * 16 + m] = d[r] + off[r + 8 * h];
        if (lane == 0) sTot[t] = off[15] + seg[15];
#else
        // Fallback: shuffle-based 256-element wave scan.
        float v[8];
        float acc = 0.0f;
        const int e0 = base + lane * 8;
#pragma unroll
        for (int k = 0; k < 8; ++k) { acc += sP[e0 + k]; v[k] = acc; }
        const float mysum = acc;
        float incl = mysum;
        for (int o = 1; o < 32; o <<= 1) {
            float u = __shfl_up(incl, o);
            if (lane >= o) incl += u;
        }
        const float pre = incl - mysum;
#pragma unroll
        for (int k = 0; k < 8; ++k) sP[e0 + k] = v[k] + pre;
        if (lane == 31) sTot[t] = incl;
#endif
    }
    __syncthreads();

    // ---- Stage 3: tile totals -> exclusive tile offsets, apply -------------
    if (tid == 0) {
        float acc = 0.0f;
        for (int t = 0; t < NTILE; ++t) { float v = sTot[t]; sTot[t] = acc; acc += v; }
    }
    __syncthreads();
    for (int e = tid; e < SLEN; e += BLOCK) sP[e + 1] += sTot[e >> 8];
    __syncthreads();

    // ---- Stage 4: stream the 9 scales from 5 prefix lookups each -----------
    // Kernel j: taps = cumsum(DEC_LO) dilated by s=2^j, scaled by 2^(-j/2):
    //   q0 on [0,s), q01 on [s,2s), q012 on [2s,3s), q0123 at 3s.
    // c[n] = g0*P[a] + g1*P[a+s] + g2*P[a+2s] + g3*P[a+3s] + g4*P[a+3s+1],
    // a = n - pad, indices clamped to [0,S] (== zero padding).
    const double q0 = -0.12940952255092145;
    const double q1 = q0 + 0.22414386804185735;
    const double q2 = q1 + 0.836516303737469;
    const double q3 = q2 + 0.48296291314469025;

    const int nlo = blockIdx.y * 8192;                       // split row output
    const int nhi = (nlo + 8192 < OUTLEN) ? nlo + 8192 : OUTLEN;
    float* orow = out + (size_t)row * (NSCALES * (size_t)OUTLEN);

    double sc = 1.0;
    for (int j = 0; j < NSCALES; ++j) {
        const int s   = 1 << j;
        const int pad = (3 * s) >> 1;                        // == (K-1)//2 all j
        const float g0 = (float)(-q0 * sc);
        const float g1 = (float)((q0 - q1) * sc);
        const float g2 = (float)((q1 - q2) * sc);
        const float g3 = (float)((q2 - q3) * sc);
        const float g4 = (float)(q3 * sc);
        float* oj = orow + (size_t)j * OUTLEN;
        for (int n = nlo + tid; n < nhi; n += BLOCK) {
            const int a = n - pad;
            const float p0 = sP[clampP(a)];
            const float p1 = sP[clampP(a + s)];
            const float p2 = sP[clampP(a + 2 * s)];
            const float p3 = sP[clampP(a + 3 * s)];
            const float p4 = sP[clampP(a + 3 * s + 1)];
            const float c = fmaf(g0, p0, fmaf(g1, p1, fmaf(g2, p2,
                            fmaf(g3, p3, g4 * p4))));
            __builtin_nontemporal_store(c, oj + n);          // 151 MB stream
        }
        sc *= 0.70710678118654752440;                        // 1/sqrt(2)
    }
}

extern "C" void kernel_launch(void* const* d_in, const int* in_sizes, int n_in,
                              void* d_out, int out_size, void* d_ws, size_t ws_size,
                              hipStream_t stream) {
    (void)n_in; (void)out_size; (void)d_ws; (void)ws_size;
    const float* x = (const float*)d_in[0];
    float* out = (float*)d_out;
    const int rows = in_sizes[0] / SLEN;     // 256 rows of 16384
    dim3 grid(rows, 2, 1);                   // y: output-range split per row
    atrous_prefix_kernel<<<grid, BLOCK, 0, stream>>>(x, out);
}